// RotaryPositionalEncoding_56178172231979
// MI455X (gfx1250) — compile-verified
//
#include <hip/hip_runtime.h>
#include <math.h>

// RoPE: out[..., 2k]   =  x[2k]*cos(m*theta_k) + x[2k+1]*sin(m*theta_k)
//       out[..., 2k+1] = -x[2k]*sin(m*theta_k) + x[2k+1]*cos(m*theta_k)
// theta_k = 10000^(-2*(k//2)/1024)  -> pairs 2j and 2j+1 share theta index j,
// so each contiguous float4 (two pairs) needs exactly ONE sincos.

#define D_MODEL 1024
#define MAX_LEN 8192
#define F4_PER_ROW (D_MODEL / 4)   // 256 float4 per token row

typedef float f4 __attribute__((ext_vector_type(4)));

// Precompute the 256 distinct thetas once (accurate powf; cost is irrelevant).
// theta[j] = 10000^(-j/512), j in [0, 256)
__global__ void rope_theta_kernel(float* __restrict__ theta) {
    const int j = threadIdx.x;               // one block of 256 threads
    theta[j] = powf(10000.0f, -(float)j * (1.0f / 512.0f));
}

// One block (256 threads = 8 wave32) per token row. Thread j handles float4 j.
__global__ void __launch_bounds__(256)
rope_kernel(const float* __restrict__ x,
            const int* __restrict__ positions,
            float* __restrict__ out,
            const float* __restrict__ theta,   // 256 entries, or nullptr
            int n_rows) {
    const int row = blockIdx.x;
    if (row >= n_rows) return;
    const int j = threadIdx.x;               // 0..255 float4 index within row

    // positions[row] is uniform across the block -> scalar load (s_load_b32)
    int p = positions[row];
    p = p < 0 ? 0 : (p > (MAX_LEN - 1) ? (MAX_LEN - 1) : p);
    const float m = (float)(p + 1);

    // theta: 1KB broadcast table, stays resident in L2 (regular temporal load)
    float th;
    if (theta != nullptr) {
        th = theta[j];
    } else {
        th = powf(10000.0f, -(float)j * (1.0f / 512.0f));  // ws-too-small fallback
    }

    const float ang = m * th;
    float s, c;
    sincosf(ang, &s, &c);                    // accurate OCML sincos (one per float4)

    const size_t idx = (size_t)row * F4_PER_ROW + (size_t)j;

    // Streaming, use-once data: non-temporal b128 load/store (TH=NT) so the
    // 268 MB stream does not evict the theta table / other L2 residents.
    const f4 v = __builtin_nontemporal_load((const f4*)x + idx);

    f4 r;
    r.x = fmaf(v.x, c, v.y * s);             //  e0*c + o0*s
    r.y = fmaf(v.y, c, -(v.x * s));          // -e0*s + o0*c
    r.z = fmaf(v.z, c, v.w * s);             //  e1*c + o1*s
    r.w = fmaf(v.w, c, -(v.z * s));          // -e1*s + o1*c

    __builtin_nontemporal_store(r, (f4*)out + idx);
}

extern "C" void kernel_launch(void* const* d_in, const int* in_sizes, int n_in,
                              void* d_out, int out_size, void* d_ws, size_t ws_size,
                              hipStream_t stream) {
    const float* x         = (const float*)d_in[0];
    const int*   positions = (const int*)d_in[1];
    float*       out       = (float*)d_out;

    const int n_rows = in_sizes[1];          // BATCH*SEQ = 32768 token rows

    float* theta = nullptr;
    if (ws_size >= F4_PER_ROW * sizeof(float)) {
        theta = (float*)d_ws;
        rope_theta_kernel<<<1, F4_PER_ROW, 0, stream>>>(theta);
    }

    rope_kernel<<<n_rows, 256, 0, stream>>>(x, positions, out, theta, n_rows);
}